// GraphSAGEEncoderDecoder_13099650253559
// MI455X (gfx1250) — compile-verified
//
#include <hip/hip_runtime.h>

// ---------------------------------------------------------------------------
// GraphSAGE encoder/decoder U-Net for gfx1250 (MI455X).
// f32 end-to-end. Dense linears via V_WMMA_F32_16X16X4_F32 with a
// wave-uniform fast path (pointer-stepped, unguarded loads) and a clamped
// edge path. Segment-max via monotonic-uint atomicMax (deterministic).
// Streaming kernels use 16B vector accesses. GraphNorm uses fixed-slot
// two-stage deterministic reductions.
// ---------------------------------------------------------------------------

typedef float v2f __attribute__((ext_vector_type(2)));
typedef float v8f __attribute__((ext_vector_type(8)));

#define ORD_NEG_INF 0x007FFFFFu  // ford(-inf)
#define ORD_POS_INF 0xFF800000u  // ford(+inf)
#define GN_BLOCKS 64
#define SLOPE 0.02f

__device__ __forceinline__ unsigned ford(float f) {
  unsigned u = __float_as_uint(f);
  return (u & 0x80000000u) ? ~u : (u | 0x80000000u);
}
__device__ __forceinline__ float funord(unsigned u) {
  unsigned v = (u & 0x80000000u) ? (u & 0x7FFFFFFFu) : ~u;
  return __uint_as_float(v);
}

// ----------------------------- fills ---------------------------------------

__global__ void k_fill_u32(unsigned* __restrict__ p, unsigned v, int n) {
  int n4 = n >> 2;
  uint4 vv; vv.x = v; vv.y = v; vv.z = v; vv.w = v;
  int gid = blockIdx.x * blockDim.x + threadIdx.x;
  int gs = gridDim.x * blockDim.x;
  for (int i = gid; i < n4; i += gs) ((uint4*)p)[i] = vv;
  for (int t = (n4 << 2) + gid; t < n; t += gs) p[t] = v;  // tail (rare)
}

// --------------------- edge gather + segment max ----------------------------

// scalar path (F not divisible by 4, i.e. the F=6 input conv)
__global__ void k_edge_segmax(const float* __restrict__ x,
                              const int* __restrict__ src,
                              const int* __restrict__ dst,
                              unsigned* __restrict__ agg, int E, int F) {
  int total = E * F;
  int gs = gridDim.x * blockDim.x;
  for (int i = blockIdx.x * blockDim.x + threadIdx.x; i < total; i += gs) {
    int e = i / F;
    int f = i - e * F;
    float v = x[(size_t)src[e] * F + f];
    atomicMax(&agg[(size_t)dst[e] * F + f], ford(v));
  }
}

// vector path: 16B gathers (F % 4 == 0)
__global__ void k_edge_segmax4(const float* __restrict__ x,
                               const int* __restrict__ src,
                               const int* __restrict__ dst,
                               unsigned* __restrict__ agg, int E, int Fq,
                               int F) {
  int total = E * Fq;
  int gs = gridDim.x * blockDim.x;
  for (int i = blockIdx.x * blockDim.x + threadIdx.x; i < total; i += gs) {
    int e = i / Fq;
    int f4 = (i - e * Fq) << 2;
    float4 v = *(const float4*)(x + (size_t)src[e] * F + f4);
    unsigned* ap = agg + (size_t)dst[e] * F + f4;
    atomicMax(ap + 0, ford(v.x));
    atomicMax(ap + 1, ford(v.y));
    atomicMax(ap + 2, ford(v.z));
    atomicMax(ap + 3, ford(v.w));
  }
}

// decode in place; untouched (empty) segments -> 0.0
__global__ void k_agg_fin(unsigned* __restrict__ agg, int n) {
  float* fa = (float*)agg;
  int n4 = n >> 2;
  int gid = blockIdx.x * blockDim.x + threadIdx.x;
  int gs = gridDim.x * blockDim.x;
  for (int i = gid; i < n4; i += gs) {
    uint4 u = ((const uint4*)agg)[i];
    float4 o;
    o.x = (u.x == ORD_NEG_INF) ? 0.f : funord(u.x);
    o.y = (u.y == ORD_NEG_INF) ? 0.f : funord(u.y);
    o.z = (u.z == ORD_NEG_INF) ? 0.f : funord(u.z);
    o.w = (u.w == ORD_NEG_INF) ? 0.f : funord(u.w);
    ((float4*)fa)[i] = o;
  }
  for (int t = (n4 << 2) + gid; t < n; t += gs) {
    unsigned u = agg[t];
    fa[t] = (u == ORD_NEG_INF) ? 0.f : funord(u);
  }
}

// --------------------------- WMMA GEMM --------------------------------------
// out[N,Fo] = agg[N,K] @ wl[K,Fo] + xr[N,K] @ wr[K,Fo] + bias
// One wave per 16x16 tile; V_WMMA_F32_16X16X4_F32.
// Fast path (wave-uniform: full row tile, K%4==0): pointer-stepped loads,
// no guards, A pair as one 8B load. Edge path: clamped branch-free selects.
__global__ __launch_bounds__(256) void k_sage_gemm(
    const float* __restrict__ agg, const float* __restrict__ xr,
    const float* __restrict__ wl, const float* __restrict__ wr,
    const float* __restrict__ bias, float* __restrict__ out, int N, int K,
    int Fo) {
  int wave = (blockIdx.x * blockDim.x + threadIdx.x) >> 5;
  int lane = threadIdx.x & 31;
  int tn_cnt = Fo >> 4;                 // Fo is a multiple of 16
  int tm_cnt = (N + 15) >> 4;
  if (wave >= tm_cnt * tn_cnt) return;  // wave-uniform
  int tm = wave / tn_cnt;
  int tn = wave - tm * tn_cnt;
  int half = lane >> 4;                 // lanes 0-15: K=k0,k0+1; 16-31: k0+2,k0+3
  int l16 = lane & 15;
  int row = tm * 16 + l16;              // A-matrix M index
  int col = tn * 16 + l16;              // B-matrix N index (< Fo always)

  v8f acc = {0.f, 0.f, 0.f, 0.f, 0.f, 0.f, 0.f, 0.f};

  if (((K & 3) == 0) & (tm * 16 + 16 <= N)) {
    // ---- fast path: wave-uniform, no per-lane guards ----
    const v2f* __restrict__ ap = (const v2f*)(agg + (size_t)row * K + (half << 1));
    const float* __restrict__ bp = wl + (size_t)(half << 1) * Fo + col;
    size_t bstep = (size_t)Fo << 2;     // 4 rows of B per k-step
#pragma unroll 4
    for (int k0 = 0; k0 < K; k0 += 4) {
      v2f a = *ap;
      v2f b;
      b.x = bp[0];
      b.y = bp[Fo];
      ap += 2;                          // 4 floats
      bp += bstep;
      acc = __builtin_amdgcn_wmma_f32_16x16x4_f32(false, a, false, b,
                                                  (short)0, acc, false, false);
    }
    ap = (const v2f*)(xr + (size_t)row * K + (half << 1));
    bp = wr + (size_t)(half << 1) * Fo + col;
#pragma unroll 4
    for (int k0 = 0; k0 < K; k0 += 4) {
      v2f a = *ap;
      v2f b;
      b.x = bp[0];
      b.y = bp[Fo];
      ap += 2;
      bp += bstep;
      acc = __builtin_amdgcn_wmma_f32_16x16x4_f32(false, a, false, b,
                                                  (short)0, acc, false, false);
    }
  } else {
    // ---- edge path: clamped indices, branch-free selects ----
    bool rowok = row < N;
    int rowc = rowok ? row : N - 1;
    const float* __restrict__ Aa = agg + (size_t)rowc * K;
    const float* __restrict__ Ax = xr + (size_t)rowc * K;
    for (int pass = 0; pass < 2; ++pass) {
      const float* __restrict__ Ar = pass ? Ax : Aa;
      const float* __restrict__ W = pass ? wr : wl;
      for (int k0 = 0; k0 < K; k0 += 4) {
        int ka = k0 + half * 2;
        int kb = ka + 1;
        int kac = ka < K ? ka : 0;
        int kbc = kb < K ? kb : 0;
        float ax = Ar[kac];
        float ay = Ar[kbc];
        float bx = W[(size_t)kac * Fo + col];
        float by = W[(size_t)kbc * Fo + col];
        v2f a, b;
        a.x = (rowok && ka < K) ? ax : 0.f;
        a.y = (rowok && kb < K) ? ay : 0.f;
        b.x = (ka < K) ? bx : 0.f;
        b.y = (kb < K) ? by : 0.f;
        acc = __builtin_amdgcn_wmma_f32_16x16x4_f32(
            false, a, false, b, (short)0, acc, false, false);
      }
    }
  }

  float bb = bias[col];
#pragma unroll
  for (int r = 0; r < 8; ++r) {         // C layout: VGPR r = rows r / r+8
    int orow = tm * 16 + half * 8 + r;
    if (orow < N) out[(size_t)orow * Fo + col] = acc[r] + bb;
  }
}

// ------------------------ GraphNorm (deterministic) -------------------------
// Stage 1: fixed-slot partials. Block threads split into RG row-subgroups of
// Fq=F/4 vector-columns each (RG = 256/Fq divides exactly for F=64..512).
__global__ __launch_bounds__(256) void k_gn_stats1(const float* __restrict__ x,
                                                   float* __restrict__ psum,
                                                   float* __restrict__ psumsq,
                                                   int N, int F) {
  int Fq = F >> 2;
  int vc = threadIdx.x % Fq;
  int rg = threadIdx.x / Fq;
  int RG = blockDim.x / Fq;
  if (rg >= RG) return;
  int stride = gridDim.x * RG;
  float4 s = {0.f, 0.f, 0.f, 0.f};
  float4 q = {0.f, 0.f, 0.f, 0.f};
  for (int r = blockIdx.x * RG + rg; r < N; r += stride) {
    float4 v = *(const float4*)(x + (size_t)r * F + (vc << 2));
    s.x += v.x; s.y += v.y; s.z += v.z; s.w += v.w;
    q.x += v.x * v.x; q.y += v.y * v.y; q.z += v.z * v.z; q.w += v.w * v.w;
  }
  size_t pidx = (size_t)(blockIdx.x * RG + rg) * F + (vc << 2);
  *(float4*)(psum + pidx) = s;
  *(float4*)(psumsq + pidx) = q;
}

// Stage 2: fold partials in fixed order; produce mean*ms and w/std
__global__ void k_gn_fin(const float* __restrict__ psum,
                         const float* __restrict__ psumsq, int nb,
                         const float* __restrict__ w,
                         const float* __restrict__ ms,
                         float* __restrict__ meanms, float* __restrict__ inv,
                         int N, int F) {
  int f = blockIdx.x * blockDim.x + threadIdx.x;
  if (f >= F) return;
  float s = 0.f, q = 0.f;
  for (int i = 0; i < nb; ++i) {
    s += psum[(size_t)i * F + f];
    q += psumsq[(size_t)i * F + f];
  }
  float mean = s / (float)N;
  float ex2 = q / (float)N;
  float m = ms[f];
  float var = ex2 - 2.f * m * mean * mean + m * m * mean * mean;
  meanms[f] = mean * m;
  inv[f] = w[f] / sqrtf(var + 1e-5f);
}

// y = leaky_relu( (x - mean*ms) * (w/std) + b ), 16B vectorized
__global__ void k_gn_apply(float* __restrict__ x,
                           const float* __restrict__ meanms,
                           const float* __restrict__ inv,
                           const float* __restrict__ b, int N, int F) {
  int Fq = F >> 2;
  int total = N * Fq;
  int gs = gridDim.x * blockDim.x;
  for (int i = blockIdx.x * blockDim.x + threadIdx.x; i < total; i += gs) {
    int f4 = (i % Fq) << 2;
    float4 v = ((const float4*)x)[i];
    float4 mm = *(const float4*)(meanms + f4);
    float4 iv = *(const float4*)(inv + f4);
    float4 bb = *(const float4*)(b + f4);
    float4 y;
    y.x = (v.x - mm.x) * iv.x + bb.x;
    y.y = (v.y - mm.y) * iv.y + bb.y;
    y.z = (v.z - mm.z) * iv.z + bb.z;
    y.w = (v.w - mm.w) * iv.w + bb.w;
    y.x = y.x > 0.f ? y.x : SLOPE * y.x;
    y.y = y.y > 0.f ? y.y : SLOPE * y.y;
    y.z = y.z > 0.f ? y.z : SLOPE * y.z;
    y.w = y.w > 0.f ? y.w : SLOPE * y.w;
    ((float4*)x)[i] = y;
  }
}

// ------------------------------ pooling -------------------------------------

__global__ void k_min_reduce(const float* __restrict__ x, int n,
                             unsigned* __restrict__ minbits) {
  int n4 = n >> 2;
  int gid = blockIdx.x * blockDim.x + threadIdx.x;
  int gs = gridDim.x * blockDim.x;
  float m = __builtin_inff();
  for (int i = gid; i < n4; i += gs) {
    float4 v = ((const float4*)x)[i];
    m = fminf(m, fminf(fminf(v.x, v.y), fminf(v.z, v.w)));
  }
  for (int t = (n4 << 2) + gid; t < n; t += gs) m = fminf(m, x[t]);
  atomicMin(minbits, ford(m));  // exact min: order-independent
}

__global__ void k_pool_segmax4(const float* __restrict__ x,
                               const int* __restrict__ pm,
                               unsigned* __restrict__ agg, int Nin, int Fq,
                               int F) {
  int total = Nin * Fq;
  int gs = gridDim.x * blockDim.x;
  for (int i = blockIdx.x * blockDim.x + threadIdx.x; i < total; i += gs) {
    int r = i / Fq;
    int f4 = (i - r * Fq) << 2;
    float4 v = *(const float4*)(x + (size_t)r * F + f4);
    unsigned* ap = agg + (size_t)pm[r] * F + f4;
    atomicMax(ap + 0, ford(v.x));
    atomicMax(ap + 1, ford(v.y));
    atomicMax(ap + 2, ford(v.z));
    atomicMax(ap + 3, ford(v.w));
  }
}

__global__ void k_pool_fin(const unsigned* __restrict__ agg,
                           const unsigned* __restrict__ minbits,
                           float* __restrict__ out, int n) {
  float base = funord(*minbits) - 0.001f;
  int n4 = n >> 2;
  int gid = blockIdx.x * blockDim.x + threadIdx.x;
  int gs = gridDim.x * blockDim.x;
  for (int i = gid; i < n4; i += gs) {
    uint4 u = ((const uint4*)agg)[i];
    float4 o;
    o.x = (u.x == ORD_NEG_INF) ? base : fmaxf(funord(u.x), base);
    o.y = (u.y == ORD_NEG_INF) ? base : fmaxf(funord(u.y), base);
    o.z = (u.z == ORD_NEG_INF) ? base : fmaxf(funord(u.z), base);
    o.w = (u.w == ORD_NEG_INF) ? base : fmaxf(funord(u.w), base);
    ((float4*)out)[i] = o;
  }
  for (int t = (n4 << 2) + gid; t < n; t += gs) {
    unsigned u = agg[t];
    out[t] = (u == ORD_NEG_INF) ? base : fmaxf(funord(u), base);
  }
}

__global__ void k_unpool(const float* __restrict__ xin,
                         const int* __restrict__ pm, float* __restrict__ xout,
                         int Nout, int Fq, int F) {
  int total = Nout * Fq;
  int gs = gridDim.x * blockDim.x;
  for (int i = blockIdx.x * blockDim.x + threadIdx.x; i < total; i += gs) {
    int r = i / Fq;
    int f4 = (i - r * Fq) << 2;
    float4 v = *(const float4*)(xin + (size_t)pm[r] * F + f4);
    *(float4*)(xout + (size_t)r * F + f4) = v;
  }
}

// ------------------------------ output head ---------------------------------
// out = tanh(x @ W[64,3] + b) * 0.5
__global__ void k_out_head(const float* __restrict__ x,
                           const float* __restrict__ w,
                           const float* __restrict__ b,
                           float* __restrict__ out, int N, int F) {
  int i = blockIdx.x * blockDim.x + threadIdx.x;
  if (i >= N) return;
  float a0 = b[0], a1 = b[1], a2 = b[2];
  const float* xr = x + (size_t)i * F;
  for (int k4 = 0; k4 < F; k4 += 4) {
    float4 v = *(const float4*)(xr + k4);
    a0 += v.x * w[(k4 + 0) * 3 + 0] + v.y * w[(k4 + 1) * 3 + 0] +
          v.z * w[(k4 + 2) * 3 + 0] + v.w * w[(k4 + 3) * 3 + 0];
    a1 += v.x * w[(k4 + 0) * 3 + 1] + v.y * w[(k4 + 1) * 3 + 1] +
          v.z * w[(k4 + 2) * 3 + 1] + v.w * w[(k4 + 3) * 3 + 1];
    a2 += v.x * w[(k4 + 0) * 3 + 2] + v.y * w[(k4 + 1) * 3 + 2] +
          v.z * w[(k4 + 2) * 3 + 2] + v.w * w[(k4 + 3) * 3 + 2];
  }
  out[(size_t)i * 3 + 0] = tanhf(a0) * 0.5f;
  out[(size_t)i * 3 + 1] = tanhf(a1) * 0.5f;
  out[(size_t)i * 3 + 2] = tanhf(a2) * 0.5f;
}

// ------------------------------ host side ----------------------------------

static inline int ceil_div(int a, int b) { return (a + b - 1) / b; }
static inline int capb(int n) {
  int b = ceil_div(n, 256);
  return b > 8192 ? 8192 : (b < 1 ? 1 : b);
}

struct Scratch {
  float* A;
  float* B;
  unsigned* G;
  float* psum;
  float* psumsq;
  float* meanms;
  float* inv;
  unsigned* minbits;
};

static void run_conv(hipStream_t s, Scratch& ws, const float* xin, float* xout,
                     const int* src, const int* dst, int E, int N, int K,
                     int Fo, const float* ll, const float* lb,
                     const float* lr) {
  int nagg = N * K;
  k_fill_u32<<<capb(nagg >> 2), 256, 0, s>>>(ws.G, ORD_NEG_INF, nagg);
  if ((K & 3) == 0) {
    int Kq = K >> 2;
    k_edge_segmax4<<<capb(E * Kq), 256, 0, s>>>(xin, src, dst, ws.G, E, Kq, K);
  } else {
    k_edge_segmax<<<capb(E * K), 256, 0, s>>>(xin, src, dst, ws.G, E, K);
  }
  k_agg_fin<<<capb(nagg >> 2), 256, 0, s>>>(ws.G, nagg);
  int waves = ceil_div(N, 16) * (Fo / 16);
  k_sage_gemm<<<ceil_div(waves * 32, 256), 256, 0, s>>>(
      (const float*)ws.G, xin, ll, lr, lb, xout, N, K, Fo);
}

static void run_gnorm(hipStream_t s, Scratch& ws, float* x, int N, int F,
                      const float* w, const float* b, const float* ms) {
  int RG = 256 / (F >> 2);  // 16/8/4/2 for F=64/128/256/512
  int nb = GN_BLOCKS * RG;
  k_gn_stats1<<<GN_BLOCKS, 256, 0, s>>>(x, ws.psum, ws.psumsq, N, F);
  k_gn_fin<<<ceil_div(F, 256), 256, 0, s>>>(ws.psum, ws.psumsq, nb, w, ms,
                                            ws.meanms, ws.inv, N, F);
  k_gn_apply<<<capb((N * F) >> 2), 256, 0, s>>>(x, ws.meanms, ws.inv, b, N, F);
}

static void run_double(hipStream_t s, Scratch& ws, const float* xin,
                       float* xmid, float* xout, const int* src,
                       const int* dst, int E, int N, int Fi, int Fo,
                       void* const* d_in, int blk) {
  const float* p[12];
  for (int j = 0; j < 12; ++j) p[j] = (const float*)d_in[14 + 12 * blk + j];
  // c0_ll c0_lb c0_lr n0_w n0_b n0_ms c1_ll c1_lb c1_lr n1_w n1_b n1_ms
  run_conv(s, ws, xin, xmid, src, dst, E, N, Fi, Fo, p[0], p[1], p[2]);
  run_gnorm(s, ws, xmid, N, Fo, p[3], p[4], p[5]);
  run_conv(s, ws, xmid, xout, src, dst, E, N, Fo, Fo, p[6], p[7], p[8]);
  run_gnorm(s, ws, xout, N, Fo, p[9], p[10], p[11]);
}

static void run_pool(hipStream_t s, Scratch& ws, float* x, const int* pm,
                     int Nin, int nc, int F) {
  int Fq = F >> 2;
  k_fill_u32<<<1, 32, 0, s>>>(ws.minbits, ORD_POS_INF, 1);
  k_min_reduce<<<capb((Nin * F) >> 2), 256, 0, s>>>(x, Nin * F, ws.minbits);
  k_fill_u32<<<capb((nc * F) >> 2), 256, 0, s>>>(ws.G, ORD_NEG_INF, nc * F);
  k_pool_segmax4<<<capb(Nin * Fq), 256, 0, s>>>(x, pm, ws.G, Nin, Fq, F);
  k_pool_fin<<<capb((nc * F) >> 2), 256, 0, s>>>(ws.G, ws.minbits, x, nc * F);
}

extern "C" void kernel_launch(void* const* d_in, const int* in_sizes, int n_in,
                              void* d_out, int out_size, void* d_ws,
                              size_t ws_size, hipStream_t stream) {
  const float* x0 = (const float*)d_in[0];
  const int* ei = (const int*)d_in[1];
  const int* pm[4] = {(const int*)d_in[2], (const int*)d_in[3],
                      (const int*)d_in[4], (const int*)d_in[5]};
  const int* se[4] = {(const int*)d_in[6], (const int*)d_in[7],
                      (const int*)d_in[8], (const int*)d_in[9]};
  const int E0 = in_sizes[1] / 2;
  const int ES[4] = {in_sizes[6] / 2, in_sizes[7] / 2, in_sizes[8] / 2,
                     in_sizes[9] / 2};
  const int N0 = in_sizes[0] / 6;  // 100000
  const int NC[4] = {25000, 6250, 1600, 400};

  const int* e_src = ei;
  const int* e_dst = ei + E0;
  const int* s_src[4];
  const int* s_dst[4];
  for (int l = 0; l < 4; ++l) {
    s_src[l] = se[l];
    s_dst[l] = se[l] + ES[l];
  }

  // workspace layout (floats): A[6.6M] B[6.6M] G[6.6M u32] stats[~132k]
  const size_t BUF = 6600000;
  Scratch ws;
  float* base = (float*)d_ws;
  ws.A = base;
  ws.B = base + BUF;
  ws.G = (unsigned*)(base + 2 * BUF);
  float* st = base + 3 * BUF;
  ws.psum = st;                 // nb*F = GN_BLOCKS*1024 = 65536 slots
  ws.psumsq = st + 65536;
  ws.meanms = st + 131072;
  ws.inv = ws.meanms + 512;
  ws.minbits = (unsigned*)(ws.inv + 512);

  float* X = ws.A;
  float* Y = ws.B;

  // -------- encoder --------
  run_double(stream, ws, x0, Y, X, e_src, e_dst, E0, N0, 6, 64, d_in, 0);
  run_pool(stream, ws, X, pm[0], N0, NC[0], 64);
  run_double(stream, ws, X, Y, X, s_src[0], s_dst[0], ES[0], NC[0], 64, 128,
             d_in, 1);
  run_pool(stream, ws, X, pm[1], NC[0], NC[1], 128);
  run_double(stream, ws, X, Y, X, s_src[1], s_dst[1], ES[1], NC[1], 128, 256,
             d_in, 2);
  run_pool(stream, ws, X, pm[2], NC[1], NC[2], 256);
  run_double(stream, ws, X, Y, X, s_src[2], s_dst[2], ES[2], NC[2], 256, 512,
             d_in, 3);
  run_pool(stream, ws, X, pm[3], NC[2], NC[3], 512);

  // -------- bottleneck --------
  run_double(stream, ws, X, Y, X, s_src[3], s_dst[3], ES[3], NC[3], 512, 512,
             d_in, 4);

  // -------- decoder --------
  k_unpool<<<capb(NC[2] * 128), 256, 0, stream>>>(X, pm[3], Y, NC[2], 128, 512);
  { float* t = X; X = Y; Y = t; }
  run_double(stream, ws, X, Y, X, s_src[2], s_dst[2], ES[2], NC[2], 512, 256,
             d_in, 5);
  k_unpool<<<capb(NC[1] * 64), 256, 0, stream>>>(X, pm[2], Y, NC[1], 64, 256);
  { float* t = X; X = Y; Y = t; }
  run_double(stream, ws, X, Y, X, s_src[1], s_dst[1], ES[1], NC[1], 256, 128,
             d_in, 6);
  k_unpool<<<capb(NC[0] * 32), 256, 0, stream>>>(X, pm[1], Y, NC[0], 32, 128);
  { float* t = X; X = Y; Y = t; }
  run_double(stream, ws, X, Y, X, s_src[0], s_dst[0], ES[0], NC[0], 128, 64,
             d_in, 7);
  k_unpool<<<capb(N0 * 16), 256, 0, stream>>>(X, pm[0], Y, N0, 16, 64);
  { float* t = X; X = Y; Y = t; }
  run_double(stream, ws, X, Y, X, e_src, e_dst, E0, N0, 64, 64, d_in, 8);

  // -------- output head --------
  const float* out_w = (const float*)d_in[14 + 12 * 9 + 0];
  const float* out_b = (const float*)d_in[14 + 12 * 9 + 1];
  k_out_head<<<ceil_div(N0, 256), 256, 0, stream>>>(X, out_w, out_b,
                                                    (float*)d_out, N0, 64);
}